// AWLoss2D_27719718929044
// MI455X (gfx1250) — compile-verified
//
#include <hip/hip_runtime.h>
#include <hip/hip_bf16.h>
#include <math.h>

#define HH 32
#define WW 32
#define NSAMP 24
#define N 1024
#define NB 16
#define NBLK 64
#define NLAG (63*63)
#define EPS 1e-4f
#define LP_PITCH 18            // LDS panel row pitch: conflict-free, 8B-aligned v2f reads
#define LP_BYTES ((NBLK-1)*NB*LP_PITCH*4)   // 1008 rows * 18 floats * 4B = 72576 B

typedef __attribute__((ext_vector_type(2))) float v2f;
typedef __attribute__((ext_vector_type(8))) float v8f;

// ---------- inverse-gaussian penalty table T (constant prefactor cancels) ----------
__global__ void k_tpen(float* __restrict__ Tflat) {
  __shared__ float sg[N];
  __shared__ float smax[256], smin[256];
  int tid = threadIdx.x;
  float lmax = -1e30f, lmin = 1e30f;
  for (int i = tid; i < N; i += 256) {
    int hx = i >> 5, wy = i & 31;
    float x = -10.f + 20.f * (float)hx / 31.f;
    float y = -10.f + 20.f * (float)wy / 31.f;
    float mx = -10.f / 31.f;               // dx * dispx, dispx = -0.5
    float dxv = x - mx, dyv = y - mx;
    float g = expf(-0.5f * (dxv * dxv + dyv * dyv));
    sg[i] = g;
    lmax = fmaxf(lmax, g);
    lmin = fminf(lmin, g);
  }
  smax[tid] = lmax; smin[tid] = lmin;
  __syncthreads();
  for (int s = 128; s > 0; s >>= 1) {
    if (tid < s) {
      smax[tid] = fmaxf(smax[tid], smax[tid + s]);
      smin[tid] = fminf(smin[tid], smin[tid + s]);
    }
    __syncthreads();
  }
  float inv = 1.f / (smax[0] - smin[0]);
  for (int i = tid; i < N; i += 256)
    Tflat[i] = (smax[0] - sg[i]) * inv;
}

// ---------- 2-D autocorrelation of target tile: ac[dj+31, dc+31] ----------
__global__ void k_autocorr(const float* __restrict__ tgt, float* __restrict__ ac) {
  int s = blockIdx.x;
  __shared__ float sx[HH][WW];
  int tid = threadIdx.x;
  for (int i = tid; i < N; i += 256) sx[i >> 5][i & 31] = tgt[(size_t)s * N + i];
  __syncthreads();
  for (int lag = tid; lag < NLAG; lag += 256) {
    int dj = lag / 63 - 31, dc = lag % 63 - 31;
    int a0 = (-dj > 0) ? -dj : 0, a1 = (31 - dj < 31) ? 31 - dj : 31;
    int b0 = (-dc > 0) ? -dc : 0, b1 = (31 - dc < 31) ? 31 - dc : 31;
    float sum = 0.f;
    for (int a = a0; a <= a1; ++a)
      for (int b = b0; b <= b1; ++b)
        sum += sx[a][b] * sx[a + dj][b + dc];
    ac[(size_t)s * 4096 + lag] = sum;
  }
}

// ---------- b = Z^T pad(recon): cross-correlation with 15-left/top pad ----------
__global__ void k_rhs(const float* __restrict__ tgt, const float* __restrict__ rec,
                      float* __restrict__ bv) {
  int s = blockIdx.x;
  __shared__ float sx[HH][WW], sr[HH][WW];
  int tid = threadIdx.x;
  for (int i = tid; i < N; i += 256) {
    sx[i >> 5][i & 31] = tgt[(size_t)s * N + i];
    sr[i >> 5][i & 31] = rec[(size_t)s * N + i];
  }
  __syncthreads();
  for (int i = tid; i < N; i += 256) {
    int j = i >> 5, c = i & 31;
    float sum = 0.f;
    for (int a = 0; a < HH; ++a) {
      int ra = a + j - 15;
      if (ra < 0 || ra > 31) continue;
      for (int b = 0; b < WW; ++b) {
        int rb = b + c - 15;
        if (rb < 0 || rb > 31) continue;
        sum += sx[a][b] * sr[ra][rb];
      }
    }
    bv[(size_t)s * N + i] = sum;
  }
}

// ---------- materialize Toeplitz Gram matrix v = Z^T Z + eps*I ----------
__global__ void k_build(const float* __restrict__ ac, float* __restrict__ v) {
  int s = blockIdx.x;
  int i1 = blockIdx.y;
  int j1 = i1 >> 5, c1 = i1 & 31;
  const float* acs = ac + (size_t)s * 4096;
  float* row = v + (size_t)s * N * N + (size_t)i1 * N;
  for (int i2 = threadIdx.x; i2 < N; i2 += 256) {
    int j2 = i2 >> 5, c2 = i2 & 31;
    float val = acs[(j1 - j2 + 31) * 63 + (c1 - c2 + 31)];
    if (i2 == i1) val += EPS;
    row[i2] = val;
  }
}

// ---------- blocked Cholesky (WMMA f32, LDS panel, 2x2 register blocking) ----------
__global__ void __launch_bounds__(256) k_chol_solve(float* __restrict__ v,
                                                    const float* __restrict__ bv,
                                                    const float* __restrict__ Tflat,
                                                    float* __restrict__ loss) {
  extern __shared__ float lp[];              // k-panel: [row][LP_PITCH], <= 1008 rows
  int s = blockIdx.x;
  float* A = v + (size_t)s * N * N;
  __shared__ float d[NB][NB + 1];
  __shared__ float sy[N];
  __shared__ float red[256], red2[256];
  int tid  = threadIdx.x;
  int wid  = tid >> 5;       // 8 wave32 per block
  int lane = tid & 31;
  int half = lane >> 4;      // lane 16..31 -> 1
  int lm   = lane & 15;

  for (int k = 0; k < NBLK; ++k) {
    // Phase 1: 16x16 diagonal block Cholesky (tiny, sequential)
    d[tid >> 4][tid & 15] = A[(size_t)(k * NB + (tid >> 4)) * N + k * NB + (tid & 15)];
    __syncthreads();
    if (tid == 0) {
      for (int c = 0; c < NB; ++c) {
        float dg = d[c][c];
        for (int t = 0; t < c; ++t) dg -= d[c][t] * d[c][t];
        dg = sqrtf(fmaxf(dg, 1e-20f));
        d[c][c] = dg;
        float inv = 1.f / dg;
        for (int r = c + 1; r < NB; ++r) {
          float x = d[r][c];
          for (int t = 0; t < c; ++t) x -= d[r][t] * d[c][t];
          d[r][c] = x * inv;
        }
      }
    }
    __syncthreads();
    A[(size_t)(k * NB + (tid >> 4)) * N + k * NB + (tid & 15)] = d[tid >> 4][tid & 15];

    // Phase 2: panel TRSM in registers, dual-write to global (for solves) + LDS panel
    int m = NBLK - 1 - k;
    int rows = m * NB;
    for (int r = tid; r < rows; r += 256) {
      size_t gi = (size_t)((k + 1) * NB + r);
      float* arow = A + gi * N + (size_t)k * NB;
      float xr[NB];
      #pragma unroll
      for (int c = 0; c < NB; ++c) xr[c] = arow[c];
      #pragma unroll
      for (int c = 0; c < NB; ++c) {
        float x = xr[c];
        for (int t = 0; t < c; ++t) x -= xr[t] * d[c][t];
        xr[c] = x / d[c][c];
      }
      #pragma unroll
      for (int c = 0; c < NB; ++c) {
        arow[c] = xr[c];
        lp[r * LP_PITCH + c] = xr[c];
      }
    }
    __syncthreads();

    // Phase 3: trailing SYRK, 32x32 super-tile per wave iteration (2x2 blocking).
    // Each sub-tile: C -= Li*Lj^T via 4 chained V_WMMA_F32_16X16X4_F32;
    // fragments come from the LDS panel (conflict-free ds_load_b64).
    int M2 = (m + 1) >> 1;
    int np2 = M2 * (M2 + 1) / 2;
    for (int p = wid; p < np2; p += 8) {           // wave-uniform: EXEC all ones
      int R = (int)((sqrtf(8.f * (float)p + 1.f) - 1.f) * 0.5f);
      while ((R + 1) * (R + 2) / 2 <= p) ++R;
      while (R * (R + 1) / 2 > p) --R;
      int C = p - R * (R + 1) / 2;
      bool vi1 = (2 * R + 1) < m;                  // uniform guards
      bool vj1 = (2 * C + 1) < m;
      int bi0 = k + 1 + 2 * R, bj0 = k + 1 + 2 * C;
      int pi0 = 2 * R * NB + lm;                   // LDS panel rows
      int pj0 = 2 * C * NB + lm;
      float* Ct = A + (size_t)bi0 * NB * N + (size_t)bj0 * NB;

      v8f c00, c01, c10, c11;
      #pragma unroll
      for (int q = 0; q < 8; ++q) c00[q] = Ct[(size_t)(q + 8 * half) * N + lm];
      if (vj1) {
        #pragma unroll
        for (int q = 0; q < 8; ++q) c01[q] = Ct[(size_t)(q + 8 * half) * N + NB + lm];
      }
      if (vi1) {
        #pragma unroll
        for (int q = 0; q < 8; ++q) c10[q] = Ct[(size_t)(q + 8 * half + NB) * N + lm];
        if (vj1) {
          #pragma unroll
          for (int q = 0; q < 8; ++q) c11[q] = Ct[(size_t)(q + 8 * half + NB) * N + NB + lm];
        }
      }
      #pragma unroll
      for (int k0 = 0; k0 < NB; k0 += 4) {
        int col = k0 + 2 * half;                   // even -> 8B-aligned v2f in LDS
        v2f a0 = *(const v2f*)(lp + pi0 * LP_PITCH + col);
        v2f b0 = *(const v2f*)(lp + pj0 * LP_PITCH + col);
        a0 = -a0;                                  // fold the subtraction into A
        c00 = __builtin_amdgcn_wmma_f32_16x16x4_f32(false, a0, false, b0,
                                                    (short)0, c00, false, false);
        if (vj1) {
          v2f b1 = *(const v2f*)(lp + (pj0 + NB) * LP_PITCH + col);
          c01 = __builtin_amdgcn_wmma_f32_16x16x4_f32(false, a0, false, b1,
                                                      (short)0, c01, false, false);
          if (vi1) {
            v2f a1 = *(const v2f*)(lp + (pi0 + NB) * LP_PITCH + col);
            a1 = -a1;
            c10 = __builtin_amdgcn_wmma_f32_16x16x4_f32(false, a1, false, b0,
                                                        (short)0, c10, false, false);
            c11 = __builtin_amdgcn_wmma_f32_16x16x4_f32(false, a1, false, b1,
                                                        (short)0, c11, false, false);
          }
        } else if (vi1) {
          v2f a1 = *(const v2f*)(lp + (pi0 + NB) * LP_PITCH + col);
          a1 = -a1;
          c10 = __builtin_amdgcn_wmma_f32_16x16x4_f32(false, a1, false, b0,
                                                      (short)0, c10, false, false);
        }
      }
      #pragma unroll
      for (int q = 0; q < 8; ++q) Ct[(size_t)(q + 8 * half) * N + lm] = c00[q];
      if (vj1) {
        #pragma unroll
        for (int q = 0; q < 8; ++q) Ct[(size_t)(q + 8 * half) * N + NB + lm] = c01[q];
      }
      if (vi1) {
        #pragma unroll
        for (int q = 0; q < 8; ++q) Ct[(size_t)(q + 8 * half + NB) * N + lm] = c10[q];
        if (vj1) {
          #pragma unroll
          for (int q = 0; q < 8; ++q) Ct[(size_t)(q + 8 * half + NB) * N + NB + lm] = c11[q];
        }
      }
    }
    __syncthreads();
  }

  // Forward solve  L y = b
  for (int i = tid; i < N; i += 256) sy[i] = bv[(size_t)s * N + i];
  __syncthreads();
  for (int jb = 0; jb < NBLK; ++jb) {
    if (tid == 0) {
      for (int c = 0; c < NB; ++c) {
        int j = jb * NB + c;
        float x = sy[j];
        for (int t = 0; t < c; ++t) x -= A[(size_t)j * N + jb * NB + t] * sy[jb * NB + t];
        sy[j] = x / A[(size_t)j * N + j];
      }
    }
    __syncthreads();
    for (int i = (jb + 1) * NB + tid; i < N; i += 256) {
      float a2 = 0.f;
      for (int c = 0; c < NB; ++c) a2 += A[(size_t)i * N + jb * NB + c] * sy[jb * NB + c];
      sy[i] -= a2;
    }
    __syncthreads();
  }
  // Backward solve  L^T w = y
  for (int jb = NBLK - 1; jb >= 0; --jb) {
    if (tid == 0) {
      for (int c = NB - 1; c >= 0; --c) {
        int j = jb * NB + c;
        float x = sy[j];
        for (int t = c + 1; t < NB; ++t) x -= A[(size_t)(jb * NB + t) * N + j] * sy[jb * NB + t];
        sy[j] = x / A[(size_t)j * N + j];
      }
    }
    __syncthreads();
    for (int i = tid; i < jb * NB; i += 256) {
      float a2 = 0.f;
      for (int c = 0; c < NB; ++c) a2 += A[(size_t)(jb * NB + c) * N + i] * sy[jb * NB + c];
      sy[i] -= a2;
    }
    __syncthreads();
  }

  // loss_s = 0.5 * ||T .* w|| / ||w||
  float n1 = 0.f, n2 = 0.f;
  for (int i = tid; i < N; i += 256) {
    float wv = sy[i];
    float tw = Tflat[i] * wv;
    n1 += tw * tw;
    n2 += wv * wv;
  }
  red[tid] = n1; red2[tid] = n2;
  __syncthreads();
  for (int st = 128; st > 0; st >>= 1) {
    if (tid < st) { red[tid] += red[tid + st]; red2[tid] += red2[tid + st]; }
    __syncthreads();
  }
  if (tid == 0)
    loss[s] = 0.5f * sqrtf(red[0]) / sqrtf(fmaxf(red2[0], 1e-30f));
}

// ---------- deterministic final sum (no float atomics) ----------
__global__ void k_reduce(const float* __restrict__ loss, float* __restrict__ out) {
  if (threadIdx.x == 0) {
    float s = 0.f;
    for (int i = 0; i < NSAMP; ++i) s += loss[i];
    out[0] = s;
  }
}

extern "C" void kernel_launch(void* const* d_in, const int* in_sizes, int n_in,
                              void* d_out, int out_size, void* d_ws, size_t ws_size,
                              hipStream_t stream) {
  const float* recon  = (const float*)d_in[0];   // (8,3,32,32) -> 24 tiles
  const float* target = (const float*)d_in[1];
  float* ws = (float*)d_ws;
  size_t off = 0;
  float* v   = ws + off; off += (size_t)NSAMP * N * N;   // 100.7 MB, L2-resident
  float* ac  = ws + off; off += (size_t)NSAMP * 4096;
  float* bv  = ws + off; off += (size_t)NSAMP * N;
  float* Tf  = ws + off; off += N;
  float* lo  = ws + off; off += NSAMP;
  (void)in_sizes; (void)n_in; (void)out_size; (void)ws_size;

  hipLaunchKernelGGL(k_tpen,       dim3(1),        dim3(256), 0,        stream, Tf);
  hipLaunchKernelGGL(k_autocorr,   dim3(NSAMP),    dim3(256), 0,        stream, target, ac);
  hipLaunchKernelGGL(k_rhs,        dim3(NSAMP),    dim3(256), 0,        stream, target, recon, bv);
  hipLaunchKernelGGL(k_build,      dim3(NSAMP, N), dim3(256), 0,        stream, ac, v);
  hipLaunchKernelGGL(k_chol_solve, dim3(NSAMP),    dim3(256), LP_BYTES, stream, v, bv, Tf, lo);
  hipLaunchKernelGGL(k_reduce,     dim3(1),        dim3(64),  0,        stream, lo, (float*)d_out);
}